// TranscriptEmission_78056735637949
// MI455X (gfx1250) — compile-verified
//
#include <hip/hip_runtime.h>
#include <hip/hip_bf16.h>
#include <math.h>

// ---------------------------------------------------------------------------
// NB log-prob sum, MI455X (gfx1250, wave32).
//
// obs in {0..49}, states in {0..11}  =>  only 600 distinct per-element values.
// Pass 1: build E[s][k] table (600 floats) in d_ws.
// Pass 2: stream obs/states with b128 loads, gather from LDS LUT, accumulate,
//         reduce per-wave with V_WMMA_F32_16X16X4_F32 (A = partials, B = 1s),
//         write one partial per block.
// Pass 3: single block reduces partials -> d_out[0]. Fully deterministic.
// ---------------------------------------------------------------------------

typedef __attribute__((ext_vector_type(2))) float v2f;
typedef __attribute__((ext_vector_type(8))) float v8f;

#define KMAX 50          // obs value range [0, 50)
#define MAX_TAB 640      // >= 12*50, padded
#define MAIN_BLOCKS 2048
#define BLOCK 256

// Wave32 sum via WMMA f32 16x16x4:
//   A(16x4): lane L<16 : VGPR0=A[L][0]=x, VGPR1=A[L][1]=0
//            lane L>=16: VGPR0=A[L-16][2]=x, VGPR1=A[L-16][3]=0
//   B(4x16) = all ones  ->  D[m][n] = x_m + x_{m+16}  (same for every n)
//   per-lane sum of the 8 D VGPRs: lanes 0-15 get sum(rows 0-7) = lanes 0-7,16-23
//                                  lanes 16-31 get sum(rows 8-15) = lanes 8-15,24-31
//   one shfl_xor(16) add -> full wave total in every lane.
__device__ __forceinline__ float wave_sum_wmma(float x) {
    v2f a; a.x = x;    a.y = 0.0f;
    v2f b; b.x = 1.0f; b.y = 1.0f;
    v8f c = {};
    v8f d = __builtin_amdgcn_wmma_f32_16x16x4_f32(
        /*neg_a=*/false, a, /*neg_b=*/false, b,
        /*c_mod=*/(short)0, c, /*reuse_a=*/false, /*reuse_b=*/false);
    float s = d[0] + d[1] + d[2] + d[3] + d[4] + d[5] + d[6] + d[7];
    s += __shfl_xor(s, 16, 32);
    return s;
}

// Pass 1: E[s][k] for s in [0,nstates), k in [0,KMAX).
__global__ void __launch_bounds__(BLOCK)
nb_table_kernel(const float* __restrict__ means, const float* __restrict__ phis,
                float* __restrict__ table, int nstates) {
    int idx = blockIdx.x * blockDim.x + threadIdx.x;
    int total = nstates * KMAX;
    if (idx >= total) return;
    int s = idx / KMAX;
    int k = idx - s * KMAX;
    float m  = means[s];
    float p  = phis[s];
    float ld = logf(m + p);
    float log_p   = logf(m) - ld;   // log(prob)
    float log_1mp = logf(p) - ld;   // log(1 - prob)
    float kf = (float)k;
    table[idx] = m * log_1mp + kf * log_p
               + lgammaf(kf + m) - lgammaf(kf + 1.0f) - lgammaf(m);
}

// Pass 2: streaming gather-sum. 8 B/element -> HBM-bound (~134 MB @ 23.3 TB/s).
__global__ void __launch_bounds__(BLOCK)
nb_main_kernel(const float* __restrict__ obs, const int* __restrict__ states,
               const float* __restrict__ table, float* __restrict__ partials,
               int n, int tab_n) {
    __shared__ float lut[MAX_TAB];
    __shared__ float wsum[BLOCK / 32];

    for (int i = threadIdx.x; i < tab_n; i += blockDim.x) lut[i] = table[i];
    __syncthreads();

    const int nthreads = gridDim.x * blockDim.x;
    const int gid = blockIdx.x * blockDim.x + threadIdx.x;

    float acc = 0.0f;

    // Vectorized body: float4 obs (b128) + int4 states (b128) per iteration.
    const int n4 = n >> 2;
    const float4* __restrict__ obs4 = (const float4*)obs;
    const int4*   __restrict__ st4  = (const int4*)states;
    for (int i = gid; i < n4; i += nthreads) {
        float4 o = obs4[i];
        int4   s = st4[i];
        unsigned k0 = (unsigned)(int)o.x; if (k0 > KMAX - 1u) k0 = KMAX - 1u;
        unsigned k1 = (unsigned)(int)o.y; if (k1 > KMAX - 1u) k1 = KMAX - 1u;
        unsigned k2 = (unsigned)(int)o.z; if (k2 > KMAX - 1u) k2 = KMAX - 1u;
        unsigned k3 = (unsigned)(int)o.w; if (k3 > KMAX - 1u) k3 = KMAX - 1u;
        acc += lut[s.x * KMAX + (int)k0];
        acc += lut[s.y * KMAX + (int)k1];
        acc += lut[s.z * KMAX + (int)k2];
        acc += lut[s.w * KMAX + (int)k3];
    }
    // Scalar tail (no-op for n = 2^24).
    for (int i = (n4 << 2) + gid; i < n; i += nthreads) {
        unsigned k = (unsigned)(int)obs[i]; if (k > KMAX - 1u) k = KMAX - 1u;
        acc += lut[states[i] * KMAX + (int)k];
    }

    // Per-wave reduction via WMMA (uniform flow, EXEC all ones).
    float w = wave_sum_wmma(acc);
    int lane = threadIdx.x & 31;
    int wid  = threadIdx.x >> 5;
    if (lane == 0) wsum[wid] = w;
    __syncthreads();
    if (threadIdx.x == 0) {
        float b = 0.0f;
        const int nw = BLOCK / 32;
        #pragma unroll
        for (int i = 0; i < nw; ++i) b += wsum[i];
        partials[blockIdx.x] = b;
    }
}

// Pass 3: one block folds the per-block partials into d_out[0].
__global__ void __launch_bounds__(BLOCK)
nb_final_kernel(const float* __restrict__ partials, int nparts,
                float* __restrict__ out) {
    __shared__ float wsum[BLOCK / 32];
    float acc = 0.0f;
    for (int i = threadIdx.x; i < nparts; i += blockDim.x) acc += partials[i];
    float w = wave_sum_wmma(acc);
    int lane = threadIdx.x & 31;
    int wid  = threadIdx.x >> 5;
    if (lane == 0) wsum[wid] = w;
    __syncthreads();
    if (threadIdx.x == 0) {
        float t = 0.0f;
        const int nw = BLOCK / 32;
        #pragma unroll
        for (int i = 0; i < nw; ++i) t += wsum[i];
        out[0] = t;
    }
}

extern "C" void kernel_launch(void* const* d_in, const int* in_sizes, int n_in,
                              void* d_out, int out_size, void* d_ws, size_t ws_size,
                              hipStream_t stream) {
    const float* obs    = (const float*)d_in[0];
    const int*   states = (const int*)  d_in[1];
    const float* means  = (const float*)d_in[2];
    const float* phis   = (const float*)d_in[3];
    float* out = (float*)d_out;

    const int n       = in_sizes[0];
    const int nstates = in_sizes[2];
    const int tab_n   = nstates * KMAX;   // 600 for nstates=12

    // Workspace layout: [0, 4096) table floats; [4096, ...) block partials.
    float* table    = (float*)d_ws;
    float* partials = (float*)((char*)d_ws + 4096);

    nb_table_kernel<<<(tab_n + BLOCK - 1) / BLOCK, BLOCK, 0, stream>>>(
        means, phis, table, nstates);
    nb_main_kernel<<<MAIN_BLOCKS, BLOCK, 0, stream>>>(
        obs, states, table, partials, n, tab_n);
    nb_final_kernel<<<1, BLOCK, 0, stream>>>(partials, MAIN_BLOCKS, out);
}